// Output_21328807592197
// MI455X (gfx1250) — compile-verified
//
#include <hip/hip_runtime.h>
#include <stdint.h>

// Frame interpolation: out = ((1-t)*v0*warp(f0) + t*v1*warp(f1)) / ((1-t)*v0 + t*v1 + 1e-12)
// B=8, H=720, W=1280, C=3, NHWC float32. Pure gather / bandwidth-bound.
// All per-tensor offsets fit in uint32 -> 32-bit address math + saddr/voffset VMEM.

typedef float v2f __attribute__((ext_vector_type(2)));

// Bilinear sample with TFA clamping semantics:
//   fy=clamp(floor(qy),0,H-2), fx=clamp(floor(qx),0,W-2), ay/ax=clamp(frac,0,1)
// tl/tr are contiguous (24B) and bl/br are contiguous (24B) in NHWC -> b128+b64 loads.
__device__ __forceinline__ void bilinear3(const float* __restrict__ img,
                                          uint32_t img_base,  // float offset of image b
                                          float qy, float qx, int H, int W,
                                          float res[3]) {
    float fy = floorf(qy);
    float fx = floorf(qx);
    fy = fminf(fmaxf(fy, 0.0f), (float)(H - 2));
    fx = fminf(fmaxf(fx, 0.0f), (float)(W - 2));
    const float ay = fminf(fmaxf(qy - fy, 0.0f), 1.0f);
    const float ax = fminf(fmaxf(qx - fx, 0.0f), 1.0f);
    const uint32_t iy = (uint32_t)(int)fy;   // clamped -> non-negative
    const uint32_t ix = (uint32_t)(int)fx;

    const uint32_t off0 = img_base + (iy * (uint32_t)W + ix) * 3u;  // row iy
    const uint32_t off1 = off0 + (uint32_t)W * 3u;                  // row iy+1

    float t0[6], t1[6];
    __builtin_memcpy(t0, img + off0, 24);   // [tl0 tl1 tl2 tr0 tr1 tr2]
    __builtin_memcpy(t1, img + off1, 24);   // [bl0 bl1 bl2 br0 br1 br2]

#pragma unroll
    for (int c = 0; c < 3; ++c) {
        const float top = t0[c] + ax * (t0[c + 3] - t0[c]);
        const float bot = t1[c] + ax * (t1[c + 3] - t1[c]);
        res[c] = top + ay * (bot - top);
    }
}

__global__ __launch_bounds__(256) void warp_blend_kernel(
    const float* __restrict__ frame0,
    const float* __restrict__ flow0,
    const float* __restrict__ vis0,
    const float* __restrict__ frame1,
    const float* __restrict__ flow1,
    const float* __restrict__ vis1,
    const float* __restrict__ t_ptr,
    float* __restrict__ out,
    int H, int W) {
    const uint32_t j = blockIdx.x * blockDim.x + threadIdx.x;
    const uint32_t i = blockIdx.y;
    const uint32_t b = blockIdx.z;
    if (j >= (uint32_t)W) return;

    // 32-bit linear pixel index (max 7.37M) and per-image float offset.
    const uint32_t p = (b * (uint32_t)H + i) * (uint32_t)W + j;
    const uint32_t img_base = b * (uint32_t)H * (uint32_t)W * 3u;

    // Uniform scalar -> s_load_b32 (single SMEM fetch per wave).
    const float t = t_ptr[0];

    // Single-use streaming inputs: nontemporal so they don't evict frame
    // texels from the 192MB L2 (both frames = 177MB, ~fits).
    const v2f fl0 = __builtin_nontemporal_load((const v2f*)flow0 + p);  // (rowflow, colflow)
    const v2f fl1 = __builtin_nontemporal_load((const v2f*)flow1 + p);
    const float v0 = __builtin_nontemporal_load(vis0 + p);
    const float v1 = __builtin_nontemporal_load(vis1 + p);

    float w0[3], w1[3];
    bilinear3(frame0, img_base, (float)i - fl0.x, (float)j - fl0.y, H, W, w0);
    bilinear3(frame1, img_base, (float)i - fl1.x, (float)j - fl1.y, H, W, w1);

    const float a = (1.0f - t) * v0;
    const float c = t * v1;
    const float inv = 1.0f / (a + (c + 1e-12f));

    float* o = out + p * 3u;
    __builtin_nontemporal_store((a * w0[0] + c * w1[0]) * inv, o + 0);
    __builtin_nontemporal_store((a * w0[1] + c * w1[1]) * inv, o + 1);
    __builtin_nontemporal_store((a * w0[2] + c * w1[2]) * inv, o + 2);
}

extern "C" void kernel_launch(void* const* d_in, const int* in_sizes, int n_in,
                              void* d_out, int out_size, void* d_ws, size_t ws_size,
                              hipStream_t stream) {
    const float* frame0 = (const float*)d_in[0];
    const float* f_t0   = (const float*)d_in[1];
    const float* v_t0   = (const float*)d_in[2];
    const float* frame1 = (const float*)d_in[3];
    const float* f_t1   = (const float*)d_in[4];
    const float* v_t1   = (const float*)d_in[5];
    const float* t_ptr  = (const float*)d_in[6];
    float* out = (float*)d_out;

    // Shapes fixed by the reference harness: [B,720,1280,*]
    const int H = 720;
    const int W = 1280;
    const int B = in_sizes[2] / (H * W);  // v_t0 is B*H*W

    dim3 block(256, 1, 1);                       // 8 wave32 waves
    dim3 grid((W + 255) / 256, H, B);            // no integer div in-kernel
    warp_blend_kernel<<<grid, block, 0, stream>>>(frame0, f_t0, v_t0,
                                                  frame1, f_t1, v_t1,
                                                  t_ptr, out, H, W);
}